// PraxisMemory_40381282517139
// MI455X (gfx1250) — compile-verified
//
#include <hip/hip_runtime.h>
#include <math.h>

// Problem constants (from reference)
#define H_   16
#define HD   64
#define MMEM 16384
#define B_   2
#define S_   512
#define NQ   (B_ * S_)       // 1024 queries per head
#define TOPK 8

// Tiling
#define NW     4             // waves per block
#define NTHR   (NW * 32)     // 128 threads
#define QPB    (NW * 16)     // 64 queries per block
#define CH     64            // memories staged per chunk
#define NCHUNK (MMEM / CH)   // 256 chunks
#define LSTR   68            // padded floats per staged row (16B-aligned, conflict-free)

typedef __attribute__((ext_vector_type(2))) float v2f;
typedef __attribute__((ext_vector_type(8))) float v8f;

// ---- async global -> LDS copy (CDNA5 ASYNCcnt path), with sync fallback ----
#if __has_builtin(__builtin_amdgcn_global_load_async_to_lds_b128) && \
    __has_builtin(__builtin_amdgcn_global_load_async_to_lds_b32)  && \
    __has_builtin(__builtin_amdgcn_s_wait_asynccnt)
#define USE_ASYNC 1
#else
#define USE_ASYNC 0
#endif

#if USE_ASYNC
typedef int v4i_t __attribute__((vector_size(16)));
typedef __attribute__((address_space(1))) v4i_t g128_t;   // global int4*
typedef __attribute__((address_space(3))) v4i_t l128_t;   // LDS int4*
typedef __attribute__((address_space(1))) int   gi32_t;   // global int*
typedef __attribute__((address_space(3))) int   li32_t;   // LDS int*
#endif

__device__ __forceinline__ void cp16_g2l(const float* g, float* l) {
#if USE_ASYNC
    __builtin_amdgcn_global_load_async_to_lds_b128((g128_t*)g, (l128_t*)l, 0, 0);
#else
    *(float4*)l = *(const float4*)g;
#endif
}
__device__ __forceinline__ void cp4_g2l(const float* g, float* l) {
#if USE_ASYNC
    __builtin_amdgcn_global_load_async_to_lds_b32((gi32_t*)g, (li32_t*)l, 0, 0);
#else
    *l = *g;
#endif
}
__device__ __forceinline__ void async_wait() {
#if USE_ASYNC
    __builtin_amdgcn_s_wait_asynccnt(0);
#endif
}

// ---- wave32 lane broadcast via LDS-permute hardware (no LDS memory used) ----
__device__ __forceinline__ float bcast_f(float v, int srcLane) {
    return __int_as_float(__builtin_amdgcn_ds_bpermute(srcLane << 2, __float_as_int(v)));
}
__device__ __forceinline__ int bcast_i(int v, int srcLane) {
    return __builtin_amdgcn_ds_bpermute(srcLane << 2, v);
}

// Branchless sorted-descending insert into a private top-8 list.
// If s <= ts[7] this is a no-op, so it is safe to run for all lanes.
__device__ __forceinline__ void insert8(float (&ts)[TOPK], int (&ti)[TOPK], float s, int idx) {
#pragma unroll
    for (int i = 0; i < TOPK; i++) {
        bool c = s > ts[i];
        float tmps = ts[i]; int tmpi = ti[i];
        ts[i] = c ? s   : tmps;
        ti[i] = c ? idx : tmpi;
        s     = c ? tmps : s;
        idx   = c ? tmpi : idx;
    }
}

// ---------------------------------------------------------------------------
// Kernel 1: per-row inverse L2 norm, rows of 64 floats. One wave per row.
// dst[r] = 1 / max(||src[r,:]||, 1e-8)   (matches F.normalize eps semantics)
// ---------------------------------------------------------------------------
__global__ void rownorm_inv(const float* __restrict__ src, float* __restrict__ dst, int nrows) {
    int row  = (int)((blockIdx.x * blockDim.x + threadIdx.x) >> 5);
    int lane = threadIdx.x & 31;
    if (row >= nrows) return;
    const float* p = src + ((size_t)row << 6) + (lane << 1);
    float x = p[0], y = p[1];
    float ss = x * x + y * y;
#pragma unroll
    for (int off = 16; off > 0; off >>= 1)
        ss += bcast_f(ss, lane ^ off);
    if (lane == 0)
        dst[row] = 1.0f / fmaxf(sqrtf(ss), 1e-8f);
}

// ---------------------------------------------------------------------------
// Kernel 2: main. grid = H * (NQ/64) = 256 blocks, 128 threads (4 waves).
// Block = (head, 64-query tile). Key chunks of 64 rows are double-buffer
// staged in LDS via async global->LDS DMA; all 4 waves (one 16-query tile
// each) consume the same stage with V_WMMA_F32_16X16X4_F32 and keep private
// top-8 lists of normalized scores. Then gather + gated blend.
// ---------------------------------------------------------------------------
__global__ __launch_bounds__(NTHR) void praxis_main(
    const float* __restrict__ query,     // [B,H,S,64]
    const float* __restrict__ outputs,   // [B,H,S,64]
    const float* __restrict__ gate,      // [H]
    const float* __restrict__ kmem,      // [H,M,64]
    const float* __restrict__ vmem,      // [H,M,64]
    const float* __restrict__ qinv,      // [B*H*S]
    const float* __restrict__ kinv,      // [H*M]
    float* __restrict__ out)             // [B,H,S,64]
{
    const int h    = blockIdx.x >> 4;          // 16 blocks per head
    const int n0   = (blockIdx.x & 15) << 6;   // first of 64 queries
    const int tid  = threadIdx.x;
    const int w    = tid >> 5;                 // wave id 0..3 -> query tile
    const int lane = tid & 31;
    const int half = lane >> 4;                // 0: K 0-1 / rows 0-7, 1: K 2-3 / rows 8-15
    const int r15  = lane & 15;
    const int koff = half << 1;
    const int rowsel = half << 3;

    __shared__ __align__(16) float stage[2][CH * LSTR];  // 2 x 17408 B
    __shared__ float skv[2][CH];
    __shared__ float s_fs[QPB][TOPK];
    __shared__ int   s_fi[QPB][TOPK];

    // ---- preload B operand: this wave's query tile (16 queries x 64).
    // 32-bit B layout: lanes 0-15 hold (K=4kk+0,4kk+1), lanes 16-31 (K=4kk+2,4kk+3).
    const int nq = n0 + (w << 4) + r15;
    const int qb = nq >> 9, qs = nq & 511;
    const float* qrow = query + ((((size_t)qb * H_ + h) * S_ + qs) << 6) + koff;
    v2f bq[16];
#pragma unroll
    for (int kk = 0; kk < 16; kk++)
        bq[kk] = *(const v2f*)(qrow + (kk << 2));

    float ts[TOPK]; int ti[TOPK];
#pragma unroll
    for (int i = 0; i < TOPK; i++) { ts[i] = -3.0e38f; ti[i] = 0; }

    const float* kmh   = kmem + (((size_t)h * MMEM) << 6);
    const float* kinvh = kinv + h * MMEM;

    // ---- stage one 64-row key chunk: 1024 16B segments, 8 per thread + kinv
    auto copy_chunk = [&](int m0c, int buf) {
#pragma unroll
        for (int i = 0; i < 8; i++) {
            int seg = (i << 7) + tid;          // 0..1023
            int row = seg >> 4;
            int c4  = (seg & 15) << 2;
            cp16_g2l(kmh + (((size_t)(m0c + row)) << 6) + c4,
                     &stage[buf][row * LSTR + c4]);
        }
        if (tid < CH) cp4_g2l(kinvh + m0c + tid, &skv[buf][tid]);
    };

    copy_chunk(0, 0);
    async_wait();
    __syncthreads();

    for (int c = 0; c < NCHUNK; ++c) {
        const int cur = c & 1;
        const int m0c = c * CH;
        if (c + 1 < NCHUNK)
            copy_chunk(m0c + CH, cur ^ 1);     // prefetch next chunk (async DMA)

        // ---- compute: 4 subtiles of 16 memory rows from LDS
        for (int sub = 0; sub < 4; sub++) {
            // A layout: M = lane&15, lanes 0-15 hold (K=4kk+0,4kk+1), 16-31 (+2,+3).
            const float* lrow = &stage[cur][((sub << 4) + r15) * LSTR + koff];
            v8f acc0 = {}; v8f acc1 = {};
#pragma unroll
            for (int kk = 0; kk < 16; kk += 2) {
                v2f a0 = *(const v2f*)(lrow + (kk << 2));
                v2f a1 = *(const v2f*)(lrow + ((kk + 1) << 2));
                acc0 = __builtin_amdgcn_wmma_f32_16x16x4_f32(
                           false, a0, false, bq[kk],     (short)0, acc0, false, false);
                acc1 = __builtin_amdgcn_wmma_f32_16x16x4_f32(
                           false, a1, false, bq[kk + 1], (short)0, acc1, false, false);
            }
#pragma unroll
            for (int j = 0; j < 8; j++) {
                float kj  = skv[cur][(sub << 4) + rowsel + j];   // LDS broadcast
                float sc  = (acc0[j] + acc1[j]) * kj;            // key-normalized score
                int   idx = m0c + (sub << 4) + rowsel + j;
                bool need = sc > ts[TOPK - 1];
                if (__builtin_amdgcn_ballot_w32(need))           // wave-uniform fast skip
                    insert8(ts, ti, sc, idx);
            }
        }
        async_wait();       // our DMA for chunk c+1 must land before next iteration
        __syncthreads();    // and everyone must be done reading buf `cur`
    }

    // ---- merge the two half-lists (lane l <-> lane l^16) for the same query
    float os[TOPK]; int oi[TOPK];
#pragma unroll
    for (int k = 0; k < TOPK; k++) {
        os[k] = bcast_f(ts[k], lane ^ 16);
        oi[k] = bcast_i(ti[k], lane ^ 16);
    }
#pragma unroll
    for (int k = 0; k < TOPK; k++) insert8(ts, ti, os[k], oi[k]);

    if (lane < 16) {
        int b2 = nq >> 9, s2 = nq & 511;
        float qi = qinv[((size_t)b2 * H_ + h) * S_ + s2] * 0.125f;  // / sqrt(64)
        int q = (w << 4) + lane;
#pragma unroll
        for (int k = 0; k < TOPK; k++) { s_fs[q][k] = ts[k] * qi; s_fi[q][k] = ti[k]; }
    }
    __syncthreads();

    // ---- gather + weighted sum + gate blend. 8 passes x (8 queries x 16 thr x 4 dims)
    const float g = 1.0f / (1.0f + __expf(-gate[h]));
    const int dp = (tid & 15) << 2;
    for (int qq = 0; qq < 8; qq++) {
        int q  = (qq << 3) + (tid >> 4);
        int nn = n0 + q;
        int b2 = nn >> 9, s2 = nn & 511;
        float ax = 0.f, ay = 0.f, az = 0.f, aw = 0.f;
#pragma unroll
        for (int k = 0; k < TOPK; k++) {
            float scv = s_fs[q][k];
            const float* vr = vmem + ((((size_t)h * MMEM) + s_fi[q][k]) << 6) + dp;
            float4 vv = *(const float4*)vr;   // 16 lanes -> coalesced 256B row
            ax += scv * vv.x; ay += scv * vv.y; az += scv * vv.z; aw += scv * vv.w;
        }
        size_t obase = ((((size_t)b2 * H_ + h) * S_ + s2) << 6) + dp;
        float4 pv = *(const float4*)(outputs + obase);
        float4 res;
        res.x = g * ax + (1.f - g) * pv.x;
        res.y = g * ay + (1.f - g) * pv.y;
        res.z = g * az + (1.f - g) * pv.z;
        res.w = g * aw + (1.f - g) * pv.w;
        *(float4*)(out + obase) = res;
    }
}

// ---------------------------------------------------------------------------
// Launch
// ---------------------------------------------------------------------------
extern "C" void kernel_launch(void* const* d_in, const int* in_sizes, int n_in,
                              void* d_out, int out_size, void* d_ws, size_t ws_size,
                              hipStream_t stream) {
    (void)in_sizes; (void)n_in; (void)out_size; (void)ws_size;
    // setup_inputs order: 0 inputs, 1 query, 2 key, 3 value, 4 outputs,
    //                     5 gate, 6 key_memories, 7 value_memories
    const float* query   = (const float*)d_in[1];
    const float* outputs = (const float*)d_in[4];
    const float* gate    = (const float*)d_in[5];
    const float* kmem    = (const float*)d_in[6];
    const float* vmem    = (const float*)d_in[7];
    float* out  = (float*)d_out;

    float* qinv = (float*)d_ws;                       // B*H*S   = 16384 floats
    float* kinv = qinv + (size_t)B_ * H_ * S_;        // H*M     = 262144 floats

    const int qrows = B_ * H_ * S_;
    const int krows = H_ * MMEM;
    rownorm_inv<<<dim3((qrows + 7) / 8), 256, 0, stream>>>(query, qinv, qrows);
    rownorm_inv<<<dim3((krows + 7) / 8), 256, 0, stream>>>(kmem, kinv, krows);

    praxis_main<<<dim3(H_ * (NQ / 64)), NTHR, 0, stream>>>(
        query, outputs, gate, kmem, vmem, qinv, kinv, out);
}